// SpikeLoss_79456894976005
// MI455X (gfx1250) — compile-verified
//
#include <hip/hip_runtime.h>

// SpikeLoss for MI455X (gfx1250, wave32).
// Bandwidth-bound streaming reduction: 419 MB read once -> ~18 us at 23.3 TB/s.
// Per-thread serial leaky integrator over its own contiguous T=100 chain,
// non-temporal b128 loads (data > 192MB L2, stream-once), WMMA 16x16x4_f32
// used as a one-instruction 32-lane reduction tree, one f32 atomic per block.

typedef float v2f __attribute__((ext_vector_type(2)));
typedef float v4f __attribute__((ext_vector_type(4)));
typedef float v8f __attribute__((ext_vector_type(8)));

#define T_STEPS 100           // trailing axis length (fixed by reference)
#define DECAY   0.5f          // 1 - 1/tau_s, tau_s = 2
#define INV_TAU 0.5f          // 1/tau_s

__global__ void spike_loss_init(float* out) {
    if (threadIdx.x == 0 && blockIdx.x == 0) out[0] = 0.0f;
}

__global__ __launch_bounds__(256) void spike_loss_kernel(
    const float* __restrict__ outputs,
    const float* __restrict__ target,
    float* __restrict__ loss,
    int n_chains)
{
    const int chain  = blockIdx.x * blockDim.x + threadIdx.x;
    const int valid  = (chain < n_chains) ? 1 : 0;
    // Branch-free OOB handling so EXEC stays all-ones for the WMMA below.
    const long long base = valid ? (long long)chain * T_STEPS : 0ll;

    const v4f* po = (const v4f*)(outputs + base);
    const v4f* pt = (const v4f*)(target  + base);

    float syn = 0.0f;
    float acc = 0.0f;
#pragma unroll
    for (int i = 0; i < T_STEPS / 4; ++i) {
        // Stream-once data: non-temporal hint keeps the 419 MB stream from
        // evicting anything useful in L2 (global_load_b128 ... nt).
        v4f o = __builtin_nontemporal_load(po + i);
        v4f x = __builtin_nontemporal_load(pt + i);
#pragma unroll
        for (int j = 0; j < 4; ++j) {
            syn = fmaf(syn, DECAY, x[j]);           // syn = 0.5*syn + x_t
            float d = fmaf(-syn, INV_TAU, o[j]);    // d = o - syn/tau_s
            acc = fmaf(d, d, acc);                  // acc += d*d
        }
    }
    acc *= (float)valid;

    // ---- wave32 reduction via V_WMMA_F32_16X16X4_F32 ----
    // A layout (16x4 f32): lanes 0-15: a[0]->K=0, a[1]->K=1 ; lanes 16-31:
    // a[0]->K=2, a[1]->K=3. With a = {p, 0}: A[m][0] = p_m, A[m][2] = p_{m+16}.
    // B = all-ones (4x16)  =>  D[m][n] = p_m + p_{m+16} for every n.
    v2f a; a[0] = acc;  a[1] = 0.0f;
    v2f b; b[0] = 1.0f; b[1] = 1.0f;
    v8f c = {};
    c = __builtin_amdgcn_wmma_f32_16x16x4_f32(
            /*neg_a=*/false, a, /*neg_b=*/false, b,
            /*c_mod=*/(short)0, c, /*reuse_a=*/false, /*reuse_b=*/false);

    // Sum the 8 accumulator VGPRs: lanes 0-15 hold M=0..7 of their column,
    // lanes 16-31 hold M=8..15. Column sums are identical across n, so the
    // wave total is t(lane L) + t(lane L^16).
    float t = c[0] + c[1] + c[2] + c[3] + c[4] + c[5] + c[6] + c[7];
    t += __shfl_xor(t, 16, 32);

    // ---- block reduction: 8 waves -> 1 atomic per block ----
    __shared__ float wsum[8];
    const int lane = threadIdx.x & 31;
    const int wave = threadIdx.x >> 5;
    if (lane == 0) wsum[wave] = t;
    __syncthreads();
    if (threadIdx.x == 0) {
        float s = 0.0f;
#pragma unroll
        for (int w = 0; w < 8; ++w) s += wsum[w];
        atomicAdd(loss, 0.5f * s);   // reference: 0.5 * sum(delta^2)
    }
}

extern "C" void kernel_launch(void* const* d_in, const int* in_sizes, int n_in,
                              void* d_out, int out_size, void* d_ws, size_t ws_size,
                              hipStream_t stream) {
    const float* outputs = (const float*)d_in[0];
    const float* target  = (const float*)d_in[1];
    float* loss = (float*)d_out;

    const int total    = in_sizes[0];            // 52,428,800
    const int n_chains = total / T_STEPS;        // 524,288
    const int block    = 256;                    // 8 wave32 waves
    const int grid     = (n_chains + block - 1) / block;   // 2048

    hipLaunchKernelGGL(spike_loss_init, dim3(1), dim3(1), 0, stream, loss);
    hipLaunchKernelGGL(spike_loss_kernel, dim3(grid), dim3(block), 0, stream,
                       outputs, target, loss, n_chains);
}